// FeedForwardNeighbor_39298950758677
// MI455X (gfx1250) — compile-verified
//
#include <hip/hip_runtime.h>

#define N_NODES 10000
#define N_EDGES 640000
#define D       128          // MODEL_DIM
#define IN_DIM  256          // 2*D (concat)

// workspace layout (ints padded so agg is 16B aligned)
#define OFF_PAD 10004        // off needs N_NODES+1; pad to multiple of 4
#define WS_INTS (N_NODES + OFF_PAD + N_NODES + N_EDGES)   // deg, off, cursor, bucket

typedef __attribute__((ext_vector_type(2))) float v2f;
typedef __attribute__((ext_vector_type(8))) float v8f;

// ---------------------------------------------------------------------------
// Kernel 1: zero deg / off / cursor region.
// ---------------------------------------------------------------------------
__global__ void ffn_zero_kernel(int* __restrict__ p, int n) {
    int i = blockIdx.x * blockDim.x + threadIdx.x;
    if (i < n) p[i] = 0;
}

// ---------------------------------------------------------------------------
// Kernel 2: per-dst degree histogram (native u32 atomics).
// ---------------------------------------------------------------------------
__global__ void ffn_count_kernel(const int* __restrict__ dst, int* __restrict__ deg) {
    int e = blockIdx.x * blockDim.x + threadIdx.x;
    if (e < N_EDGES) atomicAdd(&deg[dst[e]], 1);
}

// ---------------------------------------------------------------------------
// Kernel 3: exclusive scan of deg -> off (single block, 256 threads).
// ---------------------------------------------------------------------------
__global__ __launch_bounds__(256)
void ffn_scan_kernel(const int* __restrict__ deg, int* __restrict__ off) {
    __shared__ int partial[256];
    const int tid = threadIdx.x;
    const int chunk = (N_NODES + 255) / 256;      // 40
    int start = tid * chunk;
    int end = start + chunk; if (end > N_NODES) end = N_NODES;
    int sum = 0;
    for (int i = start; i < end; ++i) sum += deg[i];
    partial[tid] = sum;
    __syncthreads();
    if (tid == 0) {
        int acc = 0;
        for (int i = 0; i < 256; ++i) { int v = partial[i]; partial[i] = acc; acc += v; }
    }
    __syncthreads();
    int acc = partial[tid];
    for (int i = start; i < end; ++i) { off[i] = acc; acc += deg[i]; }
    if (tid == 0) off[N_NODES] = N_EDGES;
}

// ---------------------------------------------------------------------------
// Kernel 4: bucket fill: bucket[off[d] + cursor[d]++] = src[e].
// ---------------------------------------------------------------------------
__global__ void ffn_fill_kernel(const int* __restrict__ src, const int* __restrict__ dst,
                                const int* __restrict__ off, int* __restrict__ cursor,
                                int* __restrict__ bucket) {
    int e = blockIdx.x * blockDim.x + threadIdx.x;
    if (e >= N_EDGES) return;
    int d = dst[e];
    int pos = off[d] + atomicAdd(&cursor[d], 1);
    bucket[pos] = src[e];
}

// ---------------------------------------------------------------------------
// Kernel 5: gather-sum. One wave32 per node; lane accumulates float4 (4 cols).
// agg[n] = sum over incoming edges of feat[bucket[i]]. No f32 atomics.
// ---------------------------------------------------------------------------
__global__ __launch_bounds__(256)
void ffn_gather_kernel(const float* __restrict__ feat,
                       const int*   __restrict__ off,
                       const int*   __restrict__ bucket,
                       float*       __restrict__ agg) {
    int node = blockIdx.x * (blockDim.x >> 5) + (threadIdx.x >> 5);
    if (node >= N_NODES) return;
    int lane = threadIdx.x & 31;
    int beg = off[node];
    int end = off[node + 1];

    float4 a0 = float4{0.f, 0.f, 0.f, 0.f};
    float4 a1 = float4{0.f, 0.f, 0.f, 0.f};
    int i = beg;
    for (; i + 1 < end; i += 2) {       // 2x unrolled: two row-loads in flight
        int s0 = bucket[i];
        int s1 = bucket[i + 1];
        float4 v0 = ((const float4*)(feat + (long)s0 * D))[lane];
        float4 v1 = ((const float4*)(feat + (long)s1 * D))[lane];
        a0.x += v0.x; a0.y += v0.y; a0.z += v0.z; a0.w += v0.w;
        a1.x += v1.x; a1.y += v1.y; a1.z += v1.z; a1.w += v1.w;
    }
    if (i < end) {
        int s0 = bucket[i];
        float4 v0 = ((const float4*)(feat + (long)s0 * D))[lane];
        a0.x += v0.x; a0.y += v0.y; a0.z += v0.z; a0.w += v0.w;
    }
    a0.x += a1.x; a0.y += a1.y; a0.z += a1.z; a0.w += a1.w;
    ((float4*)(agg + (long)node * D))[lane] = a0;
}

// ---------------------------------------------------------------------------
// Kernel 6: fused 3-layer MLP with f32 WMMA (V_WMMA_F32_16X16X4_F32).
// Block = 16 nodes x 128 output cols. 8 waves; wave w owns cols [16w,16w+16).
//
// WMMA 16x16x4 f32 lane mapping (ISA 7.12.2):
//   A (16x4):  row M = lane&15 ; lanes<16: K={0,1} in v[0],v[1]; lanes>=16: K={2,3}
//   B (4x16):  col N = lane&15 ; lanes<16: K={0,1};              lanes>=16: K={2,3}
//   C/D:       VGPR j -> row (j + 8*(lane>=16)), col lane&15
// ---------------------------------------------------------------------------
#define XS 258   // padded LDS row stride for x  (258 % 64 == 2 -> conflict-free)
#define HS 130   // padded LDS row stride for h

__global__ __launch_bounds__(256)
void ffn_mlp_kernel(const float* __restrict__ agg,
                    const float* __restrict__ feat,
                    const float* __restrict__ W1, const float* __restrict__ b1,
                    const float* __restrict__ W2, const float* __restrict__ b2,
                    const float* __restrict__ W3, const float* __restrict__ b3,
                    float*       __restrict__ out) {
    __shared__ float xs[16 * XS];
    __shared__ float h1[16 * HS];
    __shared__ float h2[16 * HS];

    const int t    = threadIdx.x;
    const int lane = t & 31;
    const int wv   = t >> 5;          // 0..7: output column tile
    const int r    = lane & 15;       // A-row / B-col within tile
    const int hi   = lane >> 4;       // K half selector
    const int node0 = blockIdx.x * 16;
    const int n    = wv * 16 + r;     // global output column (0..127)

    // Stage x-tile = concat(agg, feat): 16 rows x 256 cols, coalesced.
    for (int i = t; i < 16 * IN_DIM; i += 256) {
        int row = i >> 8;             // / 256
        int col = i & 255;
        float v = (col < D) ? agg[(node0 + row) * D + col]
                            : feat[(node0 + row) * D + (col - D)];
        xs[row * XS + col] = v;
    }
    __syncthreads();

    // ---- Layer 1: h1 = relu(x @ W1 + b1), K = 256 ----
    v8f acc = {};
    for (int k = 0; k < IN_DIM; k += 4) {
        const int kb = k + 2 * hi;
        float2 af = *(const float2*)&xs[r * XS + kb];
        v2f a, b;
        a[0] = af.x;  a[1] = af.y;
        b[0] = W1[kb * D + n];
        b[1] = W1[(kb + 1) * D + n];
        acc = __builtin_amdgcn_wmma_f32_16x16x4_f32(false, a, false, b,
                                                    (short)0, acc, false, false);
    }
    {
        float bias = b1[n];
        #pragma unroll
        for (int j = 0; j < 8; ++j) {
            float h = acc[j] + bias;
            h1[(j + 8 * hi) * HS + n] = (h > 0.f) ? h : 0.f;
        }
    }
    __syncthreads();

    // ---- Layer 2: h2 = relu(h1 @ W2 + b2), K = 128 ----
    acc = (v8f){};
    for (int k = 0; k < D; k += 4) {
        const int kb = k + 2 * hi;
        float2 af = *(const float2*)&h1[r * HS + kb];
        v2f a, b;
        a[0] = af.x;  a[1] = af.y;
        b[0] = W2[kb * D + n];
        b[1] = W2[(kb + 1) * D + n];
        acc = __builtin_amdgcn_wmma_f32_16x16x4_f32(false, a, false, b,
                                                    (short)0, acc, false, false);
    }
    {
        float bias = b2[n];
        #pragma unroll
        for (int j = 0; j < 8; ++j) {
            float h = acc[j] + bias;
            h2[(j + 8 * hi) * HS + n] = (h > 0.f) ? h : 0.f;
        }
    }
    __syncthreads();

    // ---- Layer 3: out = h2 @ W3 + b3, K = 128 (no relu) ----
    acc = (v8f){};
    for (int k = 0; k < D; k += 4) {
        const int kb = k + 2 * hi;
        float2 af = *(const float2*)&h2[r * HS + kb];
        v2f a, b;
        a[0] = af.x;  a[1] = af.y;
        b[0] = W3[kb * D + n];
        b[1] = W3[(kb + 1) * D + n];
        acc = __builtin_amdgcn_wmma_f32_16x16x4_f32(false, a, false, b,
                                                    (short)0, acc, false, false);
    }
    {
        float bias = b3[n];
        #pragma unroll
        for (int j = 0; j < 8; ++j) {
            out[(node0 + j + 8 * hi) * D + n] = acc[j] + bias;
        }
    }
}

// ---------------------------------------------------------------------------
extern "C" void kernel_launch(void* const* d_in, const int* in_sizes, int n_in,
                              void* d_out, int out_size, void* d_ws, size_t ws_size,
                              hipStream_t stream) {
    const float* feat = (const float*)d_in[0];
    const int*   src  = (const int*)  d_in[1];
    const int*   dst  = (const int*)  d_in[2];
    const float* W1   = (const float*)d_in[3];
    const float* b1   = (const float*)d_in[4];
    const float* W2   = (const float*)d_in[5];
    const float* b2   = (const float*)d_in[6];
    const float* W3   = (const float*)d_in[7];
    const float* b3   = (const float*)d_in[8];
    float*       out  = (float*)d_out;

    // workspace carve-up (agg ends up 16B aligned: WS_INTS % 4 == 0)
    int*   deg    = (int*)d_ws;                // N_NODES
    int*   off    = deg + N_NODES;             // N_NODES+1 (padded to OFF_PAD)
    int*   cursor = off + OFF_PAD;             // N_NODES
    int*   bucket = cursor + N_NODES;          // N_EDGES
    float* agg    = (float*)(bucket + N_EDGES);// N_NODES * D

    // 1) zero deg/off/cursor region
    int nz = N_NODES + OFF_PAD + N_NODES;
    ffn_zero_kernel<<<(nz + 255) / 256, 256, 0, stream>>>(deg, nz);

    // 2) degree histogram
    ffn_count_kernel<<<(N_EDGES + 255) / 256, 256, 0, stream>>>(dst, deg);

    // 3) exclusive scan -> CSR offsets
    ffn_scan_kernel<<<1, 256, 0, stream>>>(deg, off);

    // 4) fill buckets with source node ids
    ffn_fill_kernel<<<(N_EDGES + 255) / 256, 256, 0, stream>>>(src, dst, off, cursor, bucket);

    // 5) gather-sum: one wave per node (8 waves/block, 1250 blocks)
    ffn_gather_kernel<<<N_NODES / 8, 256, 0, stream>>>(feat, off, bucket, agg);

    // 6) fused MLP: 16 nodes per block, 625 blocks
    ffn_mlp_kernel<<<N_NODES / 16, 256, 0, stream>>>(agg, feat,
                                                     W1, b1, W2, b2, W3, b3, out);
}